// GCNBlock_70531952935093
// MI455X (gfx1250) — compile-verified
//
#include <hip/hip_runtime.h>
#include <hip/hip_bf16.h>

typedef float v2f __attribute__((ext_vector_type(2)));
typedef float v8f __attribute__((ext_vector_type(8)));

#define N_NODES 100000
#define N_EDGES 1600000
#define IN_DIM  128
#define OUT_DIM 64
#define BN_EPS  1e-5f

// ---------------------------------------------------------------------------
// Workspace layout (floats):
//   deg : [0, 131072)                       -- 100000 used
//   h   : [131072, 131072 + 6.4M)           -- X@W
//   agg : [h_end,  h_end + 6.4M)            -- scatter-add accumulator
// ---------------------------------------------------------------------------
#define DEG_OFF 0
#define H_OFF   131072LL
#define AGG_OFF (H_OFF + (long long)N_NODES * OUT_DIM)

// ---------------- kernel 0: zero a float range (grid-stride) ----------------
__global__ void gcn_zero(float* __restrict__ p, long long n) {
    long long i = (long long)blockIdx.x * blockDim.x + threadIdx.x;
    long long s = (long long)gridDim.x * blockDim.x;
    for (; i < n; i += s) p[i] = 0.0f;
}

// ---------------- kernel 1: degree via f32 atomics --------------------------
__global__ void gcn_degree(const int* __restrict__ dst, float* __restrict__ deg) {
    int e = blockIdx.x * blockDim.x + threadIdx.x;
    if (e < N_EDGES) atomicAdd(&deg[dst[e]], 1.0f);
}

// ---------------- kernel 2: h = X @ W via V_WMMA_F32_16X16X4_F32 ------------
// One wave32 per 16-row strip; 4 accumulators cover the full 64-col output.
// W is staged in LDS in K-pair-interleaved float2 layout so each B fragment
// is one aligned ds_load_b64 straight into an even VGPR pair (no shuffles):
//   sW2[(k/2)*64 + col] = { W[k][col], W[k+1][col] }
__global__ __launch_bounds__(256) void gcn_gemm_wmma(const float* __restrict__ x,
                                                     const float* __restrict__ W,
                                                     float* __restrict__ h) {
    __shared__ v2f sW2[(IN_DIM / 2) * OUT_DIM];     // 32 KB of 320 KB/WGP LDS
    {
        float* sWf = (float*)sW2;
        for (int i = threadIdx.x; i < IN_DIM * OUT_DIM; i += blockDim.x) {
            const int k = i >> 6;                   // row of W (K index)
            const int c = i & 63;                   // col of W (N index)
            sWf[(((k >> 1) * OUT_DIM + c) << 1) + (k & 1)] = W[i];
        }
    }
    __syncthreads();

    const int wave  = threadIdx.x >> 5;
    const int lane  = threadIdx.x & 31;
    const int strip = blockIdx.x * 8 + wave;        // wave-uniform guard
    if (strip >= N_NODES / 16) return;              // EXEC all-1s for WMMA

    const int m0   = strip * 16;
    const int mrow = lane & 15;                     // A: M = lane%16
    const int hk   = lane >> 4;                     // 0: K+{0,1}, 1: K+{2,3}
    const int col  = lane & 15;                     // B/D: N = lane%16

    const float* __restrict__ xrow = x + (long long)(m0 + mrow) * IN_DIM;

    v8f acc0 = (v8f)0.0f, acc1 = (v8f)0.0f, acc2 = (v8f)0.0f, acc3 = (v8f)0.0f;

    for (int k = 0; k < IN_DIM; k += 4) {
        const int kk = k + (hk << 1);
        // A fragment: 16x4 f32 (lanes 0-15: K=k,k+1 ; lanes 16-31: K=k+2,k+3)
        const v2f a = *(const v2f*)&xrow[kk];
        // B fragments: one ds_load_b64 each, pair-contiguous in LDS
        const int kp = (k >> 1) + hk;               // K-pair index
        const v2f* __restrict__ brow = &sW2[kp * OUT_DIM + col];
        const v2f b0 = brow[0];
        const v2f b1 = brow[16];
        const v2f b2 = brow[32];
        const v2f b3 = brow[48];

        acc0 = __builtin_amdgcn_wmma_f32_16x16x4_f32(false, a, false, b0, (short)0, acc0, false, false);
        acc1 = __builtin_amdgcn_wmma_f32_16x16x4_f32(false, a, false, b1, (short)0, acc1, false, false);
        acc2 = __builtin_amdgcn_wmma_f32_16x16x4_f32(false, a, false, b2, (short)0, acc2, false, false);
        acc3 = __builtin_amdgcn_wmma_f32_16x16x4_f32(false, a, false, b3, (short)0, acc3, false, false);
    }

    // D layout: VGPR v, lanes 0-15 -> M=v ; lanes 16-31 -> M=v+8 ; N=lane%16
    const int rbase = m0 + (hk << 3);
#pragma unroll
    for (int v = 0; v < 8; ++v) {
        const long long r = (long long)(rbase + v) * OUT_DIM;
        h[r + col]      = acc0[v];
        h[r + 16 + col] = acc1[v];
        h[r + 32 + col] = acc2[v];
        h[r + 48 + col] = acc3[v];
    }
}

// ---------------- kernel 3: edge gather + scatter-add -----------------------
// One wave32 per edge; lane l handles output dims 2l, 2l+1.
__global__ void gcn_scatter(const int* __restrict__ src, const int* __restrict__ dst,
                            const float* __restrict__ h, const float* __restrict__ deg,
                            float* __restrict__ agg) {
    long long tid = (long long)blockIdx.x * blockDim.x + threadIdx.x;
    long long e   = tid >> 5;
    if (e >= N_EDGES) return;
    const int lane = (int)(tid & 31);

    const int s = src[e];
    const int d = dst[e];
    const float norm = rsqrtf(deg[s] + 1.0f) * rsqrtf(deg[d] + 1.0f);

    const float2 hv = *(const float2*)&h[(long long)s * OUT_DIM + lane * 2];
    float* ap = &agg[(long long)d * OUT_DIM + lane * 2];
    atomicAdd(ap,     hv.x * norm);
    atomicAdd(ap + 1, hv.y * norm);
}

// ---------------- kernel 4: self-loop + bias + BN(eval) + ReLU --------------
__global__ void gcn_finalize(const float* __restrict__ agg, const float* __restrict__ h,
                             const float* __restrict__ deg, const float* __restrict__ b,
                             const float* __restrict__ gamma, const float* __restrict__ beta,
                             const float* __restrict__ mean, const float* __restrict__ var,
                             float* __restrict__ out) {
    long long i = (long long)blockIdx.x * blockDim.x + threadIdx.x;
    if (i >= (long long)N_NODES * OUT_DIM) return;
    const int      d = (int)(i & (OUT_DIM - 1));
    const long long n = i >> 6;
    const float dv  = deg[n] + 1.0f;                 // +1 self loop
    const float v   = agg[i] + h[i] * (1.0f / dv) + b[d];
    const float bn  = (v - mean[d]) * rsqrtf(var[d] + BN_EPS) * gamma[d] + beta[d];
    out[i] = fmaxf(bn, 0.0f);
}

// ---------------------------------------------------------------------------
extern "C" void kernel_launch(void* const* d_in, const int* in_sizes, int n_in,
                              void* d_out, int out_size, void* d_ws, size_t ws_size,
                              hipStream_t stream) {
    const float* x     = (const float*)d_in[0];
    const int*   eidx  = (const int*)d_in[1];       // [2, E] flattened
    const float* W     = (const float*)d_in[2];
    const float* b     = (const float*)d_in[3];
    const float* gamma = (const float*)d_in[4];
    const float* beta  = (const float*)d_in[5];
    const float* rmean = (const float*)d_in[6];
    const float* rvar  = (const float*)d_in[7];
    float*       out   = (float*)d_out;

    const int* src = eidx;
    const int* dst = eidx + N_EDGES;

    float* ws  = (float*)d_ws;
    float* deg = ws + DEG_OFF;
    float* h   = ws + H_OFF;
    float* agg = ws + AGG_OFF;

    // 0) zero deg and agg
    gcn_zero<<<1024, 256, 0, stream>>>(deg, (long long)N_NODES);
    gcn_zero<<<4096, 256, 0, stream>>>(agg, (long long)N_NODES * OUT_DIM);

    // 1) degree counts
    gcn_degree<<<(N_EDGES + 255) / 256, 256, 0, stream>>>(dst, deg);

    // 2) h = X @ W (WMMA): 6250 strips, 8 waves/block
    gcn_gemm_wmma<<<(N_NODES / 16 + 7) / 8, 256, 0, stream>>>(x, W, h);

    // 3) normalized gather/scatter-add: one wave per edge
    const long long scat_threads = (long long)N_EDGES * 32;
    gcn_scatter<<<(unsigned)((scat_threads + 255) / 256), 256, 0, stream>>>(src, dst, h, deg, agg);

    // 4) fused epilogue -> out
    const long long fin = (long long)N_NODES * OUT_DIM;
    gcn_finalize<<<(unsigned)((fin + 255) / 256), 256, 0, stream>>>(agg, h, deg, b, gamma, beta,
                                                                    rmean, rvar, out);
}